// IntersectionOverUnion_33526514712628
// MI455X (gfx1250) — compile-verified
//
#include <hip/hip_runtime.h>
#include <cstdint>

typedef __attribute__((ext_vector_type(4))) float v4f;

#define IOU_EPS 1e-7f
#define TILE 64   // 64x64 output tile per 256-thread block

__global__ __launch_bounds__(256)
void IntersectionOverUnion_kernel(const float* __restrict__ boxes1,
                                  const float* __restrict__ boxes2,
                                  float* __restrict__ out,
                                  int n, int m) {
    __shared__ v4f sB1[TILE];   // 64 boxes * 16B = 1 KB
    __shared__ v4f sB2[TILE];   // 1 KB

    const int tileRow = blockIdx.y * TILE;
    const int tileCol = blockIdx.x * TILE;
    const int tid = threadIdx.y * 16 + threadIdx.x;  // 0..255

    // ---- Stage box tiles into LDS with CDNA5 async global->LDS B128 copies.
    // Threads 0..63 fetch the 64 boxes1 rows, 64..127 the 64 boxes2 cols.
    if (tid < TILE) {
        int i = tileRow + tid; if (i >= n) i = n - 1;   // clamp (padding lanes only)
        const v4f* g = (const v4f*)boxes1 + i;
        uint32_t loff = (uint32_t)(uintptr_t)(void*)&sB1[tid];
        asm volatile("global_load_async_to_lds_b128 %0, %1, off"
                     :: "v"(loff), "v"(g) : "memory");
    } else if (tid < 2 * TILE) {
        int j = tileCol + (tid - TILE); if (j >= m) j = m - 1;
        const v4f* g = (const v4f*)boxes2 + j;
        uint32_t loff = (uint32_t)(uintptr_t)(void*)&sB2[tid - TILE];
        asm volatile("global_load_async_to_lds_b128 %0, %1, off"
                     :: "v"(loff), "v"(g) : "memory");
    }
#if __has_builtin(__builtin_amdgcn_s_wait_asynccnt)
    __builtin_amdgcn_s_wait_asynccnt(0);
#else
    asm volatile("s_wait_asynccnt 0" ::: "memory");
#endif
    __syncthreads();

    // ---- 4x4 register micro-tile per thread.
    const int r0 = threadIdx.y * 4;   // row within tile
    const int c0 = threadIdx.x * 4;   // col within tile

    v4f a[4], b[4];
    float areaA[4], areaB[4];
#pragma unroll
    for (int k = 0; k < 4; ++k) {
        a[k] = sB1[r0 + k];
        areaA[k] = (a[k].z - a[k].x) * (a[k].w - a[k].y);
        b[k] = sB2[c0 + k];
        areaB[k] = (b[k].z - b[k].x) * (b[k].w - b[k].y);
    }

    const int gi0 = tileRow + r0;
    const int gj0 = tileCol + c0;
    const bool fullTile = (gi0 + 3 < n) && (gj0 + 3 < m) &&
                          ((m & 3) == 0) && ((((uintptr_t)out) & 15) == 0);

#pragma unroll
    for (int k = 0; k < 4; ++k) {
        v4f r;
#pragma unroll
        for (int l = 0; l < 4; ++l) {
            float x1 = fmaxf(a[k].x, b[l].x);
            float y1 = fmaxf(a[k].y, b[l].y);
            float x2 = fminf(a[k].z, b[l].z);
            float y2 = fminf(a[k].w, b[l].w);
            float iw = fmaxf(0.0f, x2 - x1);
            float ih = fmaxf(0.0f, y2 - y1);
            float inter = iw * ih;
            float uni = areaA[k] + areaB[l] - inter;
            r[l] = __fdividef(inter, uni + IOU_EPS);
        }
        if (fullTile) {
            // 256 MB streaming output, never re-read: non-temporal B128 store.
            __builtin_nontemporal_store(r, (v4f*)(out + (size_t)(gi0 + k) * (size_t)m + gj0));
        } else if (gi0 + k < n) {
#pragma unroll
            for (int l = 0; l < 4; ++l)
                if (gj0 + l < m)
                    out[(size_t)(gi0 + k) * (size_t)m + (gj0 + l)] = r[l];
        }
    }
}

extern "C" void kernel_launch(void* const* d_in, const int* in_sizes, int n_in,
                              void* d_out, int out_size, void* d_ws, size_t ws_size,
                              hipStream_t stream) {
    const float* boxes1 = (const float*)d_in[0];
    const float* boxes2 = (const float*)d_in[1];
    float* out = (float*)d_out;
    const int n = in_sizes[0] / 4;   // 8192
    const int m = in_sizes[1] / 4;   // 8192

    dim3 block(16, 16);
    dim3 grid((m + TILE - 1) / TILE, (n + TILE - 1) / TILE);
    IntersectionOverUnion_kernel<<<grid, block, 0, stream>>>(boxes1, boxes2, out, n, m);
}